// ViSNetBlock_25314537242668
// MI455X (gfx1250) — compile-verified
//
#include <hip/hip_runtime.h>
#include <math.h>

#define NN 25000
#define NE 400000
#define HID 128
#define NRBF 50

// d_out layout (floats)
#define OFF_H   0L
#define OFF_V   3200000L
#define OFF_F   12800000L
#define OFF_ANG 64000000L
#define OFF_DIH 67200000L
#define OFF_DU  118400000L

// workspace layout (floats)
#define WS_UNIT 0L                      // 3*NE
#define WS_CW   (WS_UNIT + 3L*NE)       // NE
#define WS_DIH  (WS_CW + NE)            // NE
#define WS_ANG  (WS_DIH + NE)           // NN
#define WS_CSA  (WS_ANG + NN)           // 128
#define WS_CSD  (WS_CSA + 128)          // 128
#define WS_PMSG (WS_CSD + 128)          // 308/2 * 128 * 2 = 39424
#define WS_PVEC (WS_PMSG + 39424L)      // 64 * 256 * 2 = 32768
#define WS_PSC  (WS_PVEC + 32768L)      // 64 * 128 * 2 = 16384
#define WS_PEDG (WS_PSC + 16384L)       // 64 * 128 * 2 = 16384

#define SMLD 130   // padded LDS row stride (floats): conflict-free, 8B aligned

typedef __attribute__((ext_vector_type(2))) float v2f;
typedef __attribute__((ext_vector_type(8))) float v8f;

static __device__ inline v8f wmma4(v2f a, v2f b, v8f c) {
  // D = A(16x4) x B(4x16) + C, full f32 precision
  return __builtin_amdgcn_wmma_f32_16x16x4_f32(false, a, false, b, (short)0, c,
                                               false, false);
}

static __device__ inline void atomAddF(float* p, float v) {
  __hip_atomic_fetch_add(p, v, __ATOMIC_RELAXED, __HIP_MEMORY_SCOPE_AGENT);
}

static __device__ inline float sigmoidf(float x) {
  return 1.0f / (1.0f + __expf(-x));
}

// ---------------- init ----------------
__global__ void k_zero(float* p, int n) {
  int i = blockIdx.x * blockDim.x + threadIdx.x;
  if (i < n) p[i] = 0.0f;
}

// Re-pack W (K x N, row-major) into K/2-pair layout so a lane's B operand
// {W[k][j], W[k+1][j]} is one aligned 8-byte load. Zero-pads K -> Kpad.
__global__ void k_pairB(const float* __restrict__ W, float* __restrict__ P,
                        int K, int N, int Kpad) {
  long i = (long)blockIdx.x * blockDim.x + threadIdx.x;
  long total = (long)(Kpad / 2) * N;
  if (i >= total) return;
  int kh = (int)(i / N), j = (int)(i % N);
  int k = 2 * kh;
  float x = (k < K)     ? W[(long)k * N + j]       : 0.f;
  float y = (k + 1 < K) ? W[(long)(k + 1) * N + j] : 0.f;
  P[2 * i]     = x;
  P[2 * i + 1] = y;
}

// column sums of W_ang / W_dih (rank-1 collapse of the broadcast GEMMs)
__global__ void k_colsums(const float* __restrict__ Wang,
                          const float* __restrict__ Wdih,
                          float* __restrict__ csa, float* __restrict__ csd) {
  int j = threadIdx.x;  // 128 threads
  float sa = 0.f, sd = 0.f;
  for (int i = 0; i < HID; ++i) { sa += Wang[i * HID + j]; sd += Wdih[i * HID + j]; }
  csa[j] = sa; csd[j] = sd;
}

// ---------------- per-edge geometry ----------------
__global__ void k_edge_geom(const float* __restrict__ pos, const int* __restrict__ ei,
                            float* __restrict__ du, float* __restrict__ unit,
                            float* __restrict__ cw) {
  int e = blockIdx.x * blockDim.x + threadIdx.x;
  if (e >= NE) return;
  int r = ei[e], c = ei[NE + e];
  float dx = pos[c * 3 + 0] - pos[r * 3 + 0];
  float dy = pos[c * 3 + 1] - pos[r * 3 + 1];
  float dz = pos[c * 3 + 2] - pos[r * 3 + 2];
  float dist = sqrtf(dx * dx + dy * dy + dz * dz) + 1e-8f;
  float inv = 1.0f / dist;
  float ux = dx * inv, uy = dy * inv, uz = dz * inv;
  unit[e * 3 + 0] = ux; unit[e * 3 + 1] = uy; unit[e * 3 + 2] = uz;
  float w = (dist < 10.0f) ? 0.5f * (__cosf(3.14159265358979f * dist * 0.1f) + 1.0f) : 0.0f;
  cw[e] = w;
  atomAddF(&du[r * 3 + 0], ux);  atomAddF(&du[r * 3 + 1], uy);  atomAddF(&du[r * 3 + 2], uz);
  atomAddF(&du[c * 3 + 0], -ux); atomAddF(&du[c * 3 + 1], -uy); atomAddF(&du[c * 3 + 2], -uz);
}

// ---------------- per-node: ang + angular_info broadcast ----------------
__global__ void k_node_ang(const float* __restrict__ du, float* __restrict__ ang_out,
                           float* __restrict__ ws_ang) {
  long t = (long)blockIdx.x * blockDim.x + threadIdx.x;
  int n = (int)(t >> 5), q = (int)(t & 31);
  if (n >= NN) return;
  float x = du[n * 3 + 0], y = du[n * 3 + 1], z = du[n * 3 + 2];
  float ang = x * x + y * y + z * z;
  float4 v4 = make_float4(ang, ang, ang, ang);
  ((float4*)(ang_out + (long)n * HID))[q] = v4;
  if (q == 0) ws_ang[n] = ang;
}

// ---------------- per-edge: dih + dihedral_info broadcast ----------------
__global__ void k_edge_dih(const float* __restrict__ du, const int* __restrict__ ei,
                           const float* __restrict__ unit, float* __restrict__ dih_out,
                           float* __restrict__ ws_dih) {
  long t = (long)blockIdx.x * blockDim.x + threadIdx.x;
  int e = (int)(t >> 5), q = (int)(t & 31);
  if (e >= NE) return;
  int r = ei[e], c = ei[NE + e];
  float ux = unit[e * 3 + 0], uy = unit[e * 3 + 1], uz = unit[e * 3 + 2];
  float ix = du[r * 3 + 0], iy = du[r * 3 + 1], iz = du[r * 3 + 2];
  float jx = du[c * 3 + 0], jy = du[c * 3 + 1], jz = du[c * 3 + 2];
  float di = ix * ux + iy * uy + iz * uz;
  float dj = jx * ux + jy * uy + jz * uz;
  float wix = ix - di * ux, wiy = iy - di * uy, wiz = iz - di * uz;
  float wjx = jx - dj * ux, wjy = jy - dj * uy, wjz = jz - dj * uz;
  float dih = wix * wjx + wiy * wjy + wiz * wjz;
  float4 v4 = make_float4(dih, dih, dih, dih);
  ((float4*)(dih_out + (long)e * HID))[q] = v4;
  if (q == 0) ws_dih[e] = dih;
}

// ---------------- h_updated = h + (h@W_scalar + b) * sigmoid(ang*csa + b_ang) ----------------
__global__ void k_h_update(const float* __restrict__ h, const float* __restrict__ Psc,
                           const float* __restrict__ bsc, const float* __restrict__ bang,
                           const float* __restrict__ csa, const float* __restrict__ ang,
                           float* __restrict__ out) {
  int wave = threadIdx.x >> 5, lane = threadIdx.x & 31;
  int t = lane & 15, hi = lane >> 4;
  int r0 = blockIdx.x * 16;
  int j = wave * 16 + t;
  int nodeA = r0 + t; if (nodeA >= NN) nodeA = NN - 1;  // clamp loads, mask stores
  v8f acc = {0.f, 0.f, 0.f, 0.f, 0.f, 0.f, 0.f, 0.f};
  for (int k0 = 0; k0 < HID; k0 += 4) {
    int kh = (k0 >> 1) + hi;
    v2f a = *(const v2f*)(h + (long)nodeA * HID + 2 * kh);
    v2f b = *(const v2f*)(Psc + 2 * ((long)kh * HID + j));
    acc = wmma4(a, b, acc);
  }
  float bs = bsc[j], ba = bang[j], cs = csa[j];
  for (int i = 0; i < 8; ++i) {
    int m = i + hi * 8, node = r0 + m;
    if (node < NN) {
      float g = sigmoidf(ang[node] * cs + ba);
      out[(long)node * HID + j] = h[(long)node * HID + j] + (acc[i] + bs) * g;
    }
  }
}

// ---------------- f_updated = f + (f@W_edge + b) * sigmoid(dih*csd + b_dih) ----------------
__global__ void k_f_update(const float* __restrict__ f, const float* __restrict__ Pedg,
                           const float* __restrict__ be, const float* __restrict__ bdih,
                           const float* __restrict__ csd, const float* __restrict__ dih,
                           float* __restrict__ out) {
  int wave = threadIdx.x >> 5, lane = threadIdx.x & 31;
  int t = lane & 15, hi = lane >> 4;
  int e0 = blockIdx.x * 16;
  int j = wave * 16 + t;
  v8f acc = {0.f, 0.f, 0.f, 0.f, 0.f, 0.f, 0.f, 0.f};
  for (int k0 = 0; k0 < HID; k0 += 4) {
    int kh = (k0 >> 1) + hi;
    v2f a = *(const v2f*)(f + (long)(e0 + t) * HID + 2 * kh);
    v2f b = *(const v2f*)(Pedg + 2 * ((long)kh * HID + j));
    acc = wmma4(a, b, acc);
  }
  float bs = be[j], bd = bdih[j], cs = csd[j];
  for (int i = 0; i < 8; ++i) {
    int m = i + hi * 8;
    long e = e0 + m;
    float g = sigmoidf(dih[e] * cs + bd);
    out[e * HID + j] = f[e * HID + j] + (acc[i] + bs) * g;
  }
}

// ---------------- fused: scalar_msg GEMM (K=306) -> LDS -> vec_weights GEMM ->
//                  vec_msg scatter-add into v_updated ----------------
__global__ void k_msg_vec(const float* __restrict__ h, const float* __restrict__ vin,
                          const float* __restrict__ rbf, const int* __restrict__ ei,
                          const float* __restrict__ Pmsg, const float* __restrict__ bmsg,
                          const float* __restrict__ Pvec, const float* __restrict__ bvec,
                          const float* __restrict__ unit, const float* __restrict__ cw,
                          float* __restrict__ vout) {
  __shared__ float sm[16 * SMLD];   // padded stride: no 16-way LDS bank conflicts
  __shared__ int srow[16], scol[16];
  int wave = threadIdx.x >> 5, lane = threadIdx.x & 31;
  int t = lane & 15, hi = lane >> 4;
  int e0 = blockIdx.x * 16;
  if (threadIdx.x < 16) {
    srow[threadIdx.x] = ei[e0 + threadIdx.x];
    scol[threadIdx.x] = ei[NE + e0 + threadIdx.x];
  }
  __syncthreads();

  // stage 1: scalar_msg tile (16 edges x 128), K = 306 (B pre-padded to 308)
  int j = wave * 16 + t;
  int eA = e0 + t;
  int colA = scol[t], rowA = srow[t];
  v8f acc = {0.f, 0.f, 0.f, 0.f, 0.f, 0.f, 0.f, 0.f};
  for (int k0 = 0; k0 < 308; k0 += 4) {
    int k = k0 + hi * 2;
    int kh = k >> 1;
    v2f a;
    if (k < 128)       a = *(const v2f*)(h + (long)colA * HID + k);
    else if (k < 256)  a = *(const v2f*)(h + (long)rowA * HID + (k - 128));
    else if (k < 306)  a = *(const v2f*)(rbf + (long)eA * NRBF + (k - 256));
    else               { a.x = 0.f; a.y = 0.f; }
    v2f b = *(const v2f*)(Pmsg + 2 * ((long)kh * HID + j));
    acc = wmma4(a, b, acc);
  }
  float bm = bmsg[j];
  for (int i = 0; i < 8; ++i) { int m = i + hi * 8; sm[m * SMLD + j] = acc[i] + bm; }
  __syncthreads();

  // stage 2: wave handles column c (w1) and c+128 (w2) of vec_weights, K=128
  int c = wave * 16 + t;
  v8f a1 = {0.f, 0.f, 0.f, 0.f, 0.f, 0.f, 0.f, 0.f};
  v8f a2 = {0.f, 0.f, 0.f, 0.f, 0.f, 0.f, 0.f, 0.f};
  for (int k0 = 0; k0 < HID; k0 += 4) {
    int kh = (k0 >> 1) + hi;
    v2f a = *(const v2f*)(sm + t * SMLD + 2 * kh);
    v2f b1 = *(const v2f*)(Pvec + 2 * ((long)kh * 256 + c));
    v2f b2 = *(const v2f*)(Pvec + 2 * ((long)kh * 256 + 128 + c));
    a1 = wmma4(a, b1, a1);
    a2 = wmma4(a, b2, a2);
  }
  float bv1 = bvec[c], bv2 = bvec[128 + c];
  for (int i = 0; i < 8; ++i) {
    int m = i + hi * 8;
    long e = e0 + m;
    float w = cw[e];
    float w1 = (a1[i] + bv1) * w;
    float w2 = (a2[i] + bv2) * w;
    int rI = srow[m], cI = scol[m];
#pragma unroll
    for (int d = 0; d < 3; ++d) {
      float u  = unit[e * 3 + d];
      float vv = vin[(long)rI * 384 + d * HID + c];
      atomAddF(&vout[(long)cI * 384 + d * HID + c], w1 * u + w2 * vv);
    }
  }
}

extern "C" void kernel_launch(void* const* d_in, const int* in_sizes, int n_in,
                              void* d_out, int out_size, void* d_ws, size_t ws_size,
                              hipStream_t stream) {
  const float* h    = (const float*)d_in[0];
  const float* v    = (const float*)d_in[1];
  const float* f    = (const float*)d_in[2];
  const float* pos  = (const float*)d_in[3];
  const float* rbf  = (const float*)d_in[4];
  const int*   ei   = (const int*)d_in[5];
  const float* Wmsg = (const float*)d_in[6];
  const float* bmsg = (const float*)d_in[7];
  const float* Wvec = (const float*)d_in[8];
  const float* bvec = (const float*)d_in[9];
  const float* Wsc  = (const float*)d_in[10];
  const float* bsc  = (const float*)d_in[11];
  const float* We   = (const float*)d_in[12];
  const float* be   = (const float*)d_in[13];
  const float* Wang = (const float*)d_in[14];
  const float* bang = (const float*)d_in[15];
  const float* Wdih = (const float*)d_in[16];
  const float* bdih = (const float*)d_in[17];

  float* out = (float*)d_out;
  float* ws  = (float*)d_ws;

  float* du      = out + OFF_DU;   // direction_units output doubles as accumulator
  float* ws_unit = ws + WS_UNIT;
  float* ws_cw   = ws + WS_CW;
  float* ws_dih  = ws + WS_DIH;
  float* ws_ang  = ws + WS_ANG;
  float* ws_csa  = ws + WS_CSA;
  float* ws_csd  = ws + WS_CSD;
  float* Pmsg    = ws + WS_PMSG;
  float* Pvec    = ws + WS_PVEC;
  float* Psc     = ws + WS_PSC;
  float* Pedg    = ws + WS_PEDG;

  // v_updated starts as copy of v (then atomic scatter-adds)
  hipMemcpyAsync(out + OFF_V, v, (size_t)NN * 3 * HID * sizeof(float),
                 hipMemcpyDeviceToDevice, stream);
  k_zero<<<(NN * 3 + 255) / 256, 256, 0, stream>>>(du, NN * 3);
  k_colsums<<<1, 128, 0, stream>>>(Wang, Wdih, ws_csa, ws_csd);

  // pre-pair weight matrices into the WMMA B-operand layout
  k_pairB<<<(154 * 128 + 255) / 256, 256, 0, stream>>>(Wmsg, Pmsg, 306, HID, 308);
  k_pairB<<<(64 * 256 + 255) / 256, 256, 0, stream>>>(Wvec, Pvec, 128, 256, 128);
  k_pairB<<<(64 * 128 + 255) / 256, 256, 0, stream>>>(Wsc, Psc, 128, HID, 128);
  k_pairB<<<(64 * 128 + 255) / 256, 256, 0, stream>>>(We, Pedg, 128, HID, 128);

  k_edge_geom<<<(NE + 255) / 256, 256, 0, stream>>>(pos, ei, du, ws_unit, ws_cw);
  k_node_ang<<<(NN * 32 + 255) / 256, 256, 0, stream>>>(du, out + OFF_ANG, ws_ang);
  k_edge_dih<<<((long)NE * 32 + 255) / 256, 256, 0, stream>>>(du, ei, ws_unit,
                                                              out + OFF_DIH, ws_dih);

  k_h_update<<<(NN + 15) / 16, 256, 0, stream>>>(h, Psc, bsc, bang, ws_csa, ws_ang,
                                                 out + OFF_H);
  k_f_update<<<NE / 16, 256, 0, stream>>>(f, Pedg, be, bdih, ws_csd, ws_dih, out + OFF_F);
  k_msg_vec<<<NE / 16, 256, 0, stream>>>(h, v, rbf, ei, Pmsg, bmsg, Pvec, bvec,
                                         ws_unit, ws_cw, out + OFF_V);
}